// Router_6992206758359
// MI455X (gfx1250) — compile-verified
//
#include <hip/hip_runtime.h>
#include <hip/hip_bf16.h>

typedef __attribute__((ext_vector_type(2))) float v2f;
typedef __attribute__((ext_vector_type(4))) float v4f;
typedef __attribute__((ext_vector_type(8))) float v8f;

#define NUM_EXPERTS 8
#define TOPK 2
#define HID 2048
#define NTOK 16384            // 4 * 4096
#define WAVES_PER_BLOCK 8     // K-split factor: one block = one token tile
#define KSLICE (HID / WAVES_PER_BLOCK)   // 256 K per wave
#define TILE 16               // tokens per tile (WMMA N dimension)

// ---------------------------------------------------------------------------
// One workgroup (8 waves) computes logits for one 16-token tile.
// Each wave runs V_WMMA_F32_16X16X4_F32 over a K-slice of 256:
//   A = gate (16 experts x 4 K, rows 8..15 forced to zero)
//   B = hidden^T (4 K x 16 tokens)
//   D: lane n (0..15) holds token n's partial logits for experts 0..7
// Partials reduced through LDS; wave 0 does softmax/top-2/aux epilogue.
// K-split x8 raises wave count to 8192 so chip-wide bytes-in-flight can
// cover the ~12 MB needed to saturate 23.3 TB/s HBM.
// ---------------------------------------------------------------------------
__global__ __launch_bounds__(32 * WAVES_PER_BLOCK)
void moe_router_main(const float* __restrict__ hidden,
                     const float* __restrict__ gate,
                     float* __restrict__ out,
                     float* __restrict__ ws) {
    __shared__ float s_part[WAVES_PER_BLOCK][TILE][NUM_EXPERTS];  // 4 KB
    __shared__ float s_acc[2 * NUM_EXPERTS];  // [0..7] counts, [8..15] prob sums

    const int tid  = threadIdx.x;
    const int wave = tid >> 5;
    const int lane = tid & 31;
    const int tile = blockIdx.x;

    if (tid < 2 * NUM_EXPERTS) s_acc[tid] = 0.0f;

    // Fragment addressing (wave32 layouts, CDNA5 ISA 7.12.2):
    //  A 16x4 f32:  lane<16 -> (M=lane,   K=k+0 in v0, k+1 in v1)
    //               lane>=16-> (M=lane-16,K=k+2 in v0, k+3 in v1)
    //  B 4x16 f32:  same pattern with N instead of M
    const int m16  = lane & 15;            // expert row (A) / token col (B)
    const int koff = (lane >> 4) << 1;     // 0 for lanes 0-15, 2 for lanes 16-31
    const int tok  = tile * TILE + m16;
    const int kbase = wave * KSLICE;
    const float emask = (m16 < NUM_EXPERTS) ? 1.0f : 0.0f;

    const float* gptr = gate   + (size_t)(m16 & 7) * HID + kbase + koff;
    const float* hptr = hidden + (size_t)tok * HID + kbase + koff;

    v8f c0 = {}, c1 = {}, c2 = {}, c3 = {};
#pragma unroll 4
    for (int k = 0; k < KSLICE; k += 16) {
        v2f a0 = *(const v2f*)(gptr + k);
        v2f a1 = *(const v2f*)(gptr + k + 4);
        v2f a2 = *(const v2f*)(gptr + k + 8);
        v2f a3 = *(const v2f*)(gptr + k + 12);
        v2f b0 = __builtin_nontemporal_load((const v2f*)(hptr + k));
        v2f b1 = __builtin_nontemporal_load((const v2f*)(hptr + k + 4));
        v2f b2 = __builtin_nontemporal_load((const v2f*)(hptr + k + 8));
        v2f b3 = __builtin_nontemporal_load((const v2f*)(hptr + k + 12));
        a0.x *= emask; a0.y *= emask;      // zero experts 8..15 (EXEC stays all-1s)
        a1.x *= emask; a1.y *= emask;
        a2.x *= emask; a2.y *= emask;
        a3.x *= emask; a3.y *= emask;
        c0 = __builtin_amdgcn_wmma_f32_16x16x4_f32(false, a0, false, b0,
                                                   (short)0, c0, false, false);
        c1 = __builtin_amdgcn_wmma_f32_16x16x4_f32(false, a1, false, b1,
                                                   (short)0, c1, false, false);
        c2 = __builtin_amdgcn_wmma_f32_16x16x4_f32(false, a2, false, b2,
                                                   (short)0, c2, false, false);
        c3 = __builtin_amdgcn_wmma_f32_16x16x4_f32(false, a3, false, b3,
                                                   (short)0, c3, false, false);
    }

    // Deposit this wave's partial logits (token lanes only).
    if (lane < TILE) {
#pragma unroll
        for (int i = 0; i < NUM_EXPERTS; ++i)
            s_part[wave][lane][i] = (c0[i] + c1[i]) + (c2[i] + c3[i]);
    }
    __syncthreads();

    // Wave 0 reduces the 8 K-slices and does the per-token epilogue.
    if (wave == 0 && lane < TILE) {
        float l[NUM_EXPERTS];
#pragma unroll
        for (int i = 0; i < NUM_EXPERTS; ++i) {
            float acc = 0.0f;
#pragma unroll
            for (int w = 0; w < WAVES_PER_BLOCK; ++w) acc += s_part[w][lane][i];
            l[i] = acc;
        }

        // softmax over 8 experts (per lane)
        float m = l[0];
#pragma unroll
        for (int i = 1; i < NUM_EXPERTS; ++i) m = fmaxf(m, l[i]);
        float p[NUM_EXPERTS], s = 0.0f;
#pragma unroll
        for (int i = 0; i < NUM_EXPERTS; ++i) { p[i] = __expf(l[i] - m); s += p[i]; }
        const float inv = 1.0f / s;
#pragma unroll
        for (int i = 0; i < NUM_EXPERTS; ++i) p[i] *= inv;

        // top-2 (first-occurrence tie break, matching lax.top_k)
        int i1 = 0; float p1 = p[0];
#pragma unroll
        for (int i = 1; i < NUM_EXPERTS; ++i) if (p[i] > p1) { p1 = p[i]; i1 = i; }
        int i2 = (i1 == 0) ? 1 : 0; float p2 = p[i2];
#pragma unroll
        for (int i = 0; i < NUM_EXPERTS; ++i)
            if (i != i1 && p[i] > p2) { p2 = p[i]; i2 = i; }

        const float wsum = p1 + p2;
        v2f w;   w.x = p1 / wsum;   w.y = p2 / wsum;
        v2f idx; idx.x = (float)i1; idx.y = (float)i2;
        *(v2f*)(out + (size_t)tok * 2)                    = w;    // routing_weights
        *(v2f*)(out + (size_t)NTOK * 2 + (size_t)tok * 2) = idx;  // selected_experts

        // aux-loss partials
        atomicAdd(&s_acc[i1], 1.0f);
        atomicAdd(&s_acc[i2], 1.0f);
#pragma unroll
        for (int i = 0; i < NUM_EXPERTS; ++i) atomicAdd(&s_acc[NUM_EXPERTS + i], p[i]);
    }
    __syncthreads();
    if (tid < 2 * NUM_EXPERTS) atomicAdd(&ws[tid], s_acc[tid]);
}

// ---------------------------------------------------------------------------
__global__ void moe_router_init(float* __restrict__ ws) {
    if (threadIdx.x < 2 * NUM_EXPERTS) ws[threadIdx.x] = 0.0f;
}

__global__ void moe_router_finalize(const float* __restrict__ ws,
                                    float* __restrict__ out) {
    if (threadIdx.x == 0) {
        float acc = 0.0f;
#pragma unroll
        for (int e = 0; e < NUM_EXPERTS; ++e) {
            const float expert_frac = ws[e] / (float)(NTOK * TOPK);
            const float router_frac = ws[NUM_EXPERTS + e] / (float)NTOK;
            acc += expert_frac * router_frac;
        }
        out[(size_t)NTOK * 2 * 2] = (float)NUM_EXPERTS * acc * 0.01f;  // aux_loss
    }
}

// ---------------------------------------------------------------------------
extern "C" void kernel_launch(void* const* d_in, const int* in_sizes, int n_in,
                              void* d_out, int out_size, void* d_ws, size_t ws_size,
                              hipStream_t stream) {
    const float* hidden = (const float*)d_in[0];  // [4,4096,2048] f32
    const float* gate   = (const float*)d_in[1];  // [8,2048] f32
    float* out = (float*)d_out;                   // 32768 w | 32768 idx | 1 aux
    float* ws  = (float*)d_ws;                    // 16 accumulators

    moe_router_init<<<1, 32, 0, stream>>>(ws);

    const int blocks = NTOK / TILE;               // 1024 tiles, one per block
    moe_router_main<<<blocks, 32 * WAVES_PER_BLOCK, 0, stream>>>(hidden, gate, out, ws);

    moe_router_finalize<<<1, 32, 0, stream>>>(ws, out);
}